// HeavySnowFault_33371895890246
// MI455X (gfx1250) — compile-verified
//
#include <hip/hip_runtime.h>
#include <hip/hip_bf16.h>

// ---------------- problem constants (match reference) ----------------
#define BATCH   8
#define CHAN    3
#define IMH     1024
#define IMW     1024
#define NSPOTS  15728
#define MASKW   (IMW / 32)        // 32 u32 words per mask row

// Gaussian 5-tap, sigma=1.5, normalized (precomputed in double, rounded)
#define G0 0.12007838f
#define G1 0.23388076f
#define G2 0.29208172f

// ---------------- tile geometry ----------------
#define TILE_W  64                // output tile width  (4 WMMA subtiles)
#define TILE_H  32                // output tile height (2 WMMA subtiles)
#define IN_W    (TILE_W + 4)      // 68: +2 halo each side
#define IN_H    (TILE_H + 4)      // 36
#define SIN_STRIDE IN_W           // 68: TDM iterate mode writes dense rows
#define SH_STRIDE  80             // 80 = 16 mod 64 -> the two lane halves of a
                                  // wave (2 H-rows apart) hit disjoint LDS banks

typedef float v2f __attribute__((ext_vector_type(2)));
typedef float v8f __attribute__((ext_vector_type(8)));

// TDM descriptor SGPR groups (probe-verified builtin arg shapes)
typedef unsigned int u32x4 __attribute__((ext_vector_type(4)));
typedef int          i32x4 __attribute__((ext_vector_type(4)));
typedef int          i32x8 __attribute__((ext_vector_type(8)));

#if defined(__has_builtin)
#  if __has_builtin(__builtin_amdgcn_tensor_load_to_lds)
#    define HAVE_TDM 1
#  endif
#endif

// Banded vertical-Gaussian matrix element A[m][k] = g[k-m] for 0<=k-m<=4 else 0.
__device__ __forceinline__ float bandw(int d) {
    float r = 0.0f;
    if (d == 0 || d == 4) r = G0;
    if (d == 1 || d == 3) r = G1;
    if (d == 2)           r = G2;
    return r;
}

// ---------------- kernel 0: zero the bit-mask ----------------
__global__ __launch_bounds__(256) void zero_mask_kernel(unsigned* __restrict__ m, int n) {
    int i = blockIdx.x * blockDim.x + threadIdx.x;
    if (i < n) m[i] = 0u;
}

// ---------------- kernel 1: scatter snow boxes into bit-mask ----------------
__global__ __launch_bounds__(256)
void snow_mask_kernel(const int* __restrict__ ys, const int* __restrict__ xs,
                      const int* __restrict__ rs, unsigned* __restrict__ mask, int n) {
    int i = blockIdx.x * blockDim.x + threadIdx.x;
    if (i >= n) return;
    int y = ys[i];
    int x = xs[i];
    int r = rs[i] + 1;                       // radius in {1,2,3}
    int b = i / NSPOTS;
    unsigned* mb = mask + (size_t)b * IMH * MASKW;

    int x0 = max(x - r, 0);
    int x1 = min(x + r, IMW - 1);
    int w0 = x0 >> 5, w1 = x1 >> 5;
    unsigned bits0, bits1 = 0u;
    if (w0 == w1) {
        bits0 = ((1u << (x1 - x0 + 1)) - 1u) << (x0 & 31);   // span <= 7 bits
    } else {
        bits0 = 0xFFFFFFFFu << (x0 & 31);
        bits1 = (1u << ((x1 & 31) + 1)) - 1u;                // (x1&31) <= 5 here
    }
    int y0 = max(y - r, 0);
    int y1 = min(y + r, IMH - 1);
    for (int yy = y0; yy <= y1; ++yy) {
        atomicOr(&mb[yy * MASKW + w0], bits0);
        if (w1 != w0) atomicOr(&mb[yy * MASKW + w1], bits1);
    }
}

// ---------------- kernel 2: fused mask-apply + separable 5x5 blur + clip ----
// Staging: Tensor Data Mover DMA (iterate mode: one 68-wide row per iteration,
// lds_addr_increment=68, global_addr_increment=1024) pulls the halo tile
// straight into LDS on TENSORcnt — no VGPR round trip. Vertical pass runs on
// the matrix pipe: Out(16x16) = A(16x20 banded) x H(20x16) as five chained
// V_WMMA_F32_16X16X4_F32 (K = 20 in chunks of 4).
__global__ __launch_bounds__(256)
void snow_blur_kernel(const float* __restrict__ x, const unsigned* __restrict__ mask,
                      float* __restrict__ out) {
    __shared__ float sIn[IN_H * SIN_STRIDE];   // raw input tile (with halo)
    __shared__ float sH [IN_H * SH_STRIDE];    // horizontally blurred rows

    const int t    = threadIdx.x;
    const int col0 = blockIdx.x * TILE_W;
    const int row0 = blockIdx.y * TILE_H;
    const int bc   = blockIdx.z;               // b*CHAN + channel
    const int b    = bc / CHAN;
    const size_t planeBase = (size_t)bc * IMH * IMW;
    const unsigned* mplane = mask + (size_t)b * IMH * MASKW;

    const int gx0 = col0 - 2, gy0 = row0 - 2;  // halo origin (may be -2)
    const int sx0 = max(gx0, 0), sy0 = max(gy0, 0);
    const int lx  = sx0 - gx0;                 // left rows/cols clipped (0 or 2)
    const int ly  = sy0 - gy0;
    const int rows = min(IN_H - ly, IMH - sy0);// valid rows to DMA

#if HAVE_TDM
    // ---- phase 1a: zero tile (covers clipped halo strips), then TDM load ----
    for (int e = t; e < IN_H * IN_W; e += 256) sIn[e] = 0.0f;
    __syncthreads();                           // zeros complete before DMA issue

    if (t < 32) {                              // one wave issues the DMA
        unsigned long long ga = (unsigned long long)(const void*)
            (x + planeBase + (size_t)sy0 * IMW + sx0);
        unsigned ldsAddr = (unsigned)(unsigned long long)(void*)sIn
                         + (unsigned)((ly * SIN_STRIDE + lx) * 4);
        int tileW   = IN_W - lx;               // 66 or 68
        int tensorW = IMW - sx0;               // right-edge OOB columns read 0

        u32x4 g0;
        g0.x = 1u;                                         // count=1 (valid D#)
        g0.y = ldsAddr;                                    // lds_addr (bytes)
        g0.z = (unsigned)(ga & 0xFFFFFFFFu);               // global_addr[31:0]
        g0.w = (unsigned)((ga >> 32) & 0x1FFFFFFu)         // global_addr[56:32]
             | 0x80000000u;                                // type=2 ("image")

        i32x8 g1;
        g1[0] = 0x000A0000;                                // data_size=4B, iterate_en
        g1[1] = (int)((unsigned)(tensorW & 0xFFFF) << 16); // tensor_dim0[15:0]
        g1[2] = (int)(((unsigned)tensorW >> 16) & 0xFFFFu) // tensor_dim0[31:16]
              | (IMH << 16);                               // tensor_dim1[15:0]
        g1[3] = (tileW << 16);                             // tile_dim0
        g1[4] = 1;                                         // tile_dim1=1 (row tiles)
        g1[5] = IMW;                                       // tensor_dim0_stride
        g1[6] = 0;
        g1[7] = 0;

        i32x4 g2;
        g2[0] = 0;                                         // tensor_dim2
        g2[1] = SIN_STRIDE;                                // lds_addr_increment (elems)
        g2[2] = IMW;                                       // global_addr_increment
        g2[3] = ((rows - 1) << 16);                        // iterate_count
        i32x4 g3 = {0, 0, 0, 0};

#if defined(__clang_major__) && __clang_major__ >= 23
        i32x8 g4 = {0, 0, 0, 0, 0, 0, 0, 0};
        __builtin_amdgcn_tensor_load_to_lds(g0, g1, g2, g3, g4, 0);
#else
        __builtin_amdgcn_tensor_load_to_lds(g0, g1, g2, g3, 0);
#endif
#if defined(__has_builtin) && __has_builtin(__builtin_amdgcn_s_wait_tensorcnt)
        __builtin_amdgcn_s_wait_tensorcnt(0);
#else
        asm volatile("s_wait_tensorcnt 0x0" ::: "memory");
#endif
    }
    __syncthreads();                           // tile visible to all waves

    // ---- phase 1b: overlay snow mask (sparse conditional LDS writes) ----
    for (int e = t; e < IN_H * IN_W; e += 256) {
        int u  = e / IN_W;
        int j  = e - u * IN_W;
        int gy = gy0 + u;
        int gx = gx0 + j;
        if (gy >= 0 && gy < IMH && gx >= 0 && gx < IMW) {
            unsigned mword = mplane[gy * MASKW + (gx >> 5)];
            if ((mword >> (gx & 31)) & 1u) sIn[u * SIN_STRIDE + j] = 0.95f;
        }
    }
#else
    // ---- fallback: VGPR staging with fused mask / zero-pad ----
    (void)lx; (void)ly; (void)rows; (void)sx0; (void)sy0;
    for (int e = t; e < IN_H * IN_W; e += 256) {
        int u  = e / IN_W;
        int j  = e - u * IN_W;
        int gy = gy0 + u;
        int gx = gx0 + j;
        float v = 0.0f;
        if (gy >= 0 && gy < IMH && gx >= 0 && gx < IMW) {
            v = x[planeBase + (size_t)gy * IMW + gx];
            unsigned mword = mplane[gy * MASKW + (gx >> 5)];
            if ((mword >> (gx & 31)) & 1u) v = 0.95f;
        }
        sIn[u * SIN_STRIDE + j] = v;
    }
#endif
    __syncthreads();

    // ---- phase 2: horizontal 5-tap (VALU, 36x64 values) ----
    for (int e = t; e < IN_H * TILE_W; e += 256) {
        int u = e >> 6;               // / 64
        int j = e & 63;
        const float* rp = &sIn[u * SIN_STRIDE + j];
        sH[u * SH_STRIDE + j] =
            G0 * (rp[0] + rp[4]) + G1 * (rp[1] + rp[3]) + G2 * rp[2];
    }
    __syncthreads();

    // ---- phase 3: vertical 5-tap via WMMA (full wave active: EXEC all-1s) ----
    const int wave = t >> 5;
    const int lane = t & 31;
    const int sy   = wave >> 2;       // 0..1  -> output rows sy*16..+15
    const int sx   = wave & 3;        // 0..3  -> output cols sx*16..+15
    const int m    = lane & 15;       // A row (lanes) / B,N column (lanes)
    const int hi   = (lane >> 4) & 1; // lane half selects K+0/K+1 vs K+2/K+3
    const int rowBaseH = sy * 16;     // H rows used: sy*16 .. sy*16+19
    const int colBase  = sx * 16;

    v8f acc = {};
#pragma unroll
    for (int c = 0; c < 5; ++c) {
        int k0 = 4 * c + hi * 2;
        v2f a, bm;
        // A (16x4 f32 chunk): VGPR0 = K {0,2}, VGPR1 = K {1,3} per lane half
        a.x = bandw(k0     - m);
        a.y = bandw(k0 + 1 - m);
        // B (4x16 f32 chunk), mirrored layout: lanes = N, lane-half picks K pair
        bm.x = sH[(rowBaseH + k0    ) * SH_STRIDE + colBase + m];
        bm.y = sH[(rowBaseH + k0 + 1) * SH_STRIDE + colBase + m];
        acc = __builtin_amdgcn_wmma_f32_16x16x4_f32(
            /*neg_a=*/false, a, /*neg_b=*/false, bm,
            /*c_mod=*/(short)0, acc, /*reuse_a=*/false, /*reuse_b=*/false);
    }

    // ---- clip + store (C/D layout: VGPR r -> rows r / r+8 per lane half) ----
#pragma unroll
    for (int r = 0; r < 8; ++r) {
        float v = fminf(fmaxf(acc[r], 0.0f), 1.0f);
        int grow = row0 + sy * 16 + r + hi * 8;
        int gcol = col0 + colBase + m;
        out[planeBase + (size_t)grow * IMW + gcol] = v;
    }
}

// ---------------- launcher ----------------
extern "C" void kernel_launch(void* const* d_in, const int* in_sizes, int n_in,
                              void* d_out, int out_size, void* d_ws, size_t ws_size,
                              hipStream_t stream) {
    const float* x  = (const float*)d_in[0];
    const int*   ys = (const int*)  d_in[1];
    const int*   xs = (const int*)  d_in[2];
    const int*   rs = (const int*)  d_in[3];
    float*       out  = (float*)d_out;
    unsigned*    mask = (unsigned*)d_ws;      // 8*1024*32 u32 = 1 MB

    const int maskWords = BATCH * IMH * MASKW;
    zero_mask_kernel<<<(maskWords + 255) / 256, 256, 0, stream>>>(mask, maskWords);

    const int nSpots = BATCH * NSPOTS;
    snow_mask_kernel<<<(nSpots + 255) / 256, 256, 0, stream>>>(ys, xs, rs, mask, nSpots);

    dim3 grid(IMW / TILE_W, IMH / TILE_H, BATCH * CHAN);
    snow_blur_kernel<<<grid, 256, 0, stream>>>(x, mask, out);
}